// Codebook2_33535104647887
// MI455X (gfx1250) — compile-verified
//
#include <hip/hip_runtime.h>

// Shapes (fixed by the reference)
#define BATCH   32
#define RLEN    512
#define NROWS   (BATCH * RLEN)      // 16384
#define IN_DIM  1024
#define D_      256
#define TWO_D   512
#define KCODES  4096
#define OUT_DIM 1024

typedef float v2f __attribute__((ext_vector_type(2)));
typedef float v8f __attribute__((ext_vector_type(8)));

// ---------------------------------------------------------------------------
// WMMA: D = A(16x4) * B(4x16) + C, fp32.
//   A: lane L<16 holds A[L][k+0..1], lane L>=16 holds A[L-16][k+2..3]
//   B: lane L<16 holds B[k+0..1][L], lane L>=16 holds B[k+2..3][L-16]
//   C/D: vgpr j, lanes 0-15 -> (M=j, N=lane), lanes 16-31 -> (M=8+j, N=lane-16)
// ---------------------------------------------------------------------------
__device__ __forceinline__ v8f wmma_f32_4(v2f a, v2f b, v8f c) {
    return __builtin_amdgcn_wmma_f32_16x16x4_f32(
        false, a, false, b, (short)0, c, false, false);
}

// ---------------------------------------------------------------------------
// CDNA5 async global->LDS copy (ASYNCcnt-tracked), GV addressing mode.
// vdst = LDS byte offset VGPR, vaddr = 64-bit global address VGPR pair.
// ---------------------------------------------------------------------------
__device__ __forceinline__ void async_ld_b128(void* lds, const void* gaddr) {
    asm volatile("global_load_async_to_lds_b128 %0, %1, off"
                 :: "v"((unsigned int)(unsigned long long)(lds)),
                    "v"((unsigned long long)gaddr)
                 : "memory");
}
__device__ __forceinline__ void wait_async0() {
    asm volatile("s_wait_asynccnt 0x0" ::: "memory");
}

// ---------------------------------------------------------------------------
// fp32 GEMM: C[M,N] = A[M,K] @ B[K,N] + bias[N]
// Block = 128 threads (4 waves), tile 64 rows x 64 cols, K-chunks of 32,
// double-buffered LDS, one barrier per chunk. Each wave: 16 rows x 64 cols
// = 4 independent WMMA accumulator chains; A-frag shared across the 4.
// A staged via async-to-LDS; B staged via register transpose-scatter so
// fragments are contiguous b64 loads (no repack movs).
// Requires M%64==0, N%64==0, K%32==0.
// ---------------------------------------------------------------------------
__global__ __launch_bounds__(128)
void gemm_bias_f32(const float* __restrict__ A, const float* __restrict__ B,
                   const float* __restrict__ bias, float* __restrict__ C,
                   int M, int N, int K) {
    __shared__ float As [2][64][36];  // 64 rows x 32 k   (144B rows: 16B aligned)
    __shared__ float BsT[2][64][36];  // 64 cols x 32 k   (col-major: frag = b64)

    const int tid  = threadIdx.x;
    const int wave = tid >> 5;
    const int lane = tid & 31;
    const int col0 = blockIdx.x * 64;
    const int row0 = blockIdx.y * 64;

    const int mrow = wave * 16 + (lane & 15);
    const int koff = (lane < 16) ? 0 : 2;
    const int ncol = lane & 15;

    const int T = K / 32;

    v8f acc0 = {}, acc1 = {}, acc2 = {}, acc3 = {};

    // ---- stage chunk 0 into buffer 0 ----
    #pragma unroll
    for (int i = 0; i < 4; ++i) {
        int idx4 = i * 128 + tid;            // A: 512 float4 (64 rows x 8 f4)
        int r = idx4 >> 3, c4 = (idx4 & 7) * 4;
        async_ld_b128(&As[0][r][c4], A + (size_t)(row0 + r) * K + c4);
    }
    #pragma unroll
    for (int i = 0; i < 4; ++i) {
        int idx4 = i * 128 + tid;            // B: 512 float4 (32 k x 16 f4)
        int r = idx4 >> 4, c4 = (idx4 & 15) * 4;
        float4 v = *(const float4*)(B + (size_t)r * N + col0 + c4);
        BsT[0][c4 + 0][r] = v.x;
        BsT[0][c4 + 1][r] = v.y;
        BsT[0][c4 + 2][r] = v.z;
        BsT[0][c4 + 3][r] = v.w;
    }
    wait_async0();
    __syncthreads();

    for (int t = 0; t < T; ++t) {
        const int p = t & 1;
        const bool has_next = (t + 1 < T);
        float4 pfb[4];

        if (has_next) {
            const int k0n = (t + 1) * 32;
            #pragma unroll
            for (int i = 0; i < 4; ++i) {
                int idx4 = i * 128 + tid;
                int r = idx4 >> 3, c4 = (idx4 & 7) * 4;
                async_ld_b128(&As[p ^ 1][r][c4],
                              A + (size_t)(row0 + r) * K + k0n + c4);
            }
            #pragma unroll
            for (int i = 0; i < 4; ++i) {
                int idx4 = i * 128 + tid;
                int r = idx4 >> 4, c4 = (idx4 & 15) * 4;
                pfb[i] = *(const float4*)(B + (size_t)(k0n + r) * N + col0 + c4);
            }
        }

        #pragma unroll
        for (int kk = 0; kk < 32; kk += 4) {
            v2f a  = *(const v2f*)&As [p][mrow][kk + koff];
            v2f b0 = *(const v2f*)&BsT[p][ncol     ][kk + koff];
            v2f b1 = *(const v2f*)&BsT[p][16 + ncol][kk + koff];
            v2f b2 = *(const v2f*)&BsT[p][32 + ncol][kk + koff];
            v2f b3 = *(const v2f*)&BsT[p][48 + ncol][kk + koff];
            acc0 = wmma_f32_4(a, b0, acc0);
            acc1 = wmma_f32_4(a, b1, acc1);
            acc2 = wmma_f32_4(a, b2, acc2);
            acc3 = wmma_f32_4(a, b3, acc3);
        }

        if (has_next) {
            #pragma unroll
            for (int i = 0; i < 4; ++i) {
                int idx4 = i * 128 + tid;
                int r = idx4 >> 4, c4 = (idx4 & 15) * 4;
                BsT[p ^ 1][c4 + 0][r] = pfb[i].x;
                BsT[p ^ 1][c4 + 1][r] = pfb[i].y;
                BsT[p ^ 1][c4 + 2][r] = pfb[i].z;
                BsT[p ^ 1][c4 + 3][r] = pfb[i].w;
            }
        }
        wait_async0();
        __syncthreads();
    }

    const int rbase = row0 + wave * 16 + ((lane < 16) ? 0 : 8);
    #pragma unroll
    for (int cb = 0; cb < 4; ++cb) {
        const int outc = col0 + cb * 16 + ncol;
        const float bv = bias ? bias[outc] : 0.0f;
        const v8f* a = (cb == 0) ? &acc0 : (cb == 1) ? &acc1 : (cb == 2) ? &acc2 : &acc3;
        #pragma unroll
        for (int j = 0; j < 8; ++j)
            C[(size_t)(rbase + j) * N + outc] = (*a)[j] + bv;
    }
}

// ---------------------------------------------------------------------------
// Codebook squared norms
// ---------------------------------------------------------------------------
__global__ __launch_bounds__(256)
void enorm_kernel(const float* __restrict__ emb1, const float* __restrict__ emb2,
                  float* __restrict__ enorm) {
    int i = blockIdx.x * 256 + threadIdx.x;
    const float* e = (i < KCODES) ? (emb1 + (size_t)i * D_)
                                  : (emb2 + (size_t)(i - KCODES) * D_);
    float s = 0.f;
    #pragma unroll 4
    for (int k = 0; k < D_; k += 4) {
        float4 v = *(const float4*)(e + k);
        s += v.x * v.x + v.y * v.y + v.z * v.z + v.w * v.w;
    }
    enorm[i] = s;
}

// ---------------------------------------------------------------------------
// VQ argmin. Block = 128 threads (4 waves) = 64 rows. Each wave holds its
// 16 z-rows (K=256) in registers (64 v2f = 128 VGPRs) and computes
// 16 rows x 64 codes per group with 4 accumulator chains. Codebook tiles
// (64 codes x 64 k) stream through double-buffered LDS via async-to-LDS.
// argmin on d = ||e||^2 - 2*score (||z||^2 is row-constant).
// ---------------------------------------------------------------------------
__global__ __launch_bounds__(128)
void vq_argmin_kernel(const float* __restrict__ h,
                      const float* __restrict__ emb1, const float* __restrict__ emb2,
                      const float* __restrict__ enorm,
                      int* __restrict__ idx1, int* __restrict__ idx2) {
    const int half = blockIdx.y;
    const float* emb = half ? emb2 : emb1;
    const float* en  = enorm + half * KCODES;
    int* idx_out     = half ? idx2 : idx1;

    __shared__ float Es[2][64][68];   // 64 codes x 64 k (272B rows: 16B aligned)
    __shared__ float redv[64][16];
    __shared__ int   redi[64][16];

    const int tid  = threadIdx.x;
    const int wave = tid >> 5;
    const int lane = tid & 31;
    const int row0 = blockIdx.x * 64 + wave * 16;

    const int koff = (lane < 16) ? 0 : 2;
    const int ncol = lane & 15;

    // A fragments in registers: 64 K-steps for this wave's 16 rows.
    v2f a_reg[64];
    {
        const float* base = h + (size_t)(row0 + (lane & 15)) * TWO_D + half * D_ + koff;
        #pragma unroll
        for (int ks = 0; ks < 64; ++ks)
            a_reg[ks] = *(const v2f*)(base + ks * 4);
    }

    float minv[8];
    int   mini[8];
    #pragma unroll
    for (int j = 0; j < 8; ++j) { minv[j] = 3.4e38f; mini[j] = 0; }

    // ---- stage chunk 0 (group 0, k-chunk 0) via async ----
    #pragma unroll
    for (int i = 0; i < 8; ++i) {
        int idx4 = i * 128 + tid;            // 1024 float4 (64 codes x 16 f4)
        int r = idx4 >> 4, c4 = (idx4 & 15) * 4;
        async_ld_b128(&Es[0][r][c4], emb + (size_t)r * D_ + c4);
    }
    wait_async0();
    __syncthreads();

    int t = 0;                                // global chunk index: g*4 + c
    for (int g = 0; g < KCODES / 64; ++g) {
        v8f acc0 = {}, acc1 = {}, acc2 = {}, acc3 = {};

        #pragma unroll
        for (int c = 0; c < 4; ++c, ++t) {
            const int p = t & 1;
            const bool has_next = (t + 1 < (KCODES / 64) * 4);

            if (has_next) {
                const int tn = t + 1;
                const int gn = tn >> 2, cn = tn & 3;
                #pragma unroll
                for (int i = 0; i < 8; ++i) {
                    int idx4 = i * 128 + tid;
                    int r = idx4 >> 4, c4 = (idx4 & 15) * 4;
                    async_ld_b128(&Es[p ^ 1][r][c4],
                                  emb + (size_t)(gn * 64 + r) * D_ + cn * 64 + c4);
                }
            }

            #pragma unroll
            for (int kk16 = 0; kk16 < 16; ++kk16) {
                v2f a = a_reg[c * 16 + kk16];
                const int kk = kk16 * 4 + koff;
                v2f b0 = *(const v2f*)&Es[p][ncol     ][kk];
                v2f b1 = *(const v2f*)&Es[p][16 + ncol][kk];
                v2f b2 = *(const v2f*)&Es[p][32 + ncol][kk];
                v2f b3 = *(const v2f*)&Es[p][48 + ncol][kk];
                acc0 = wmma_f32_4(a, b0, acc0);
                acc1 = wmma_f32_4(a, b1, acc1);
                acc2 = wmma_f32_4(a, b2, acc2);
                acc3 = wmma_f32_4(a, b3, acc3);
            }

            wait_async0();
            __syncthreads();
        }

        // epilogue: fold ||e||^2, update running per-lane argmin
        #pragma unroll
        for (int cb = 0; cb < 4; ++cb) {
            const int code = g * 64 + cb * 16 + ncol;
            const float e2 = en[code];
            const v8f* a = (cb == 0) ? &acc0 : (cb == 1) ? &acc1 : (cb == 2) ? &acc2 : &acc3;
            #pragma unroll
            for (int j = 0; j < 8; ++j) {
                float d = e2 - 2.0f * (*a)[j];
                if (d < minv[j]) { minv[j] = d; mini[j] = code; }  // strict <: first min
            }
        }
    }

    // cross-lane reduction: each local row has 16 lane-candidates
    #pragma unroll
    for (int j = 0; j < 8; ++j) {
        int r = wave * 16 + j + ((lane < 16) ? 0 : 8);
        redv[r][ncol] = minv[j];
        redi[r][ncol] = mini[j];
    }
    __syncthreads();
    if (tid < 64) {
        float bv = redv[tid][0];
        int   bi = redi[tid][0];
        #pragma unroll
        for (int q = 1; q < 16; ++q) {
            float v = redv[tid][q]; int c = redi[tid][q];
            if (v < bv || (v == bv && c < bi)) { bv = v; bi = c; }  // tie -> lowest idx
        }
        idx_out[blockIdx.x * 64 + tid] = bi;
    }
}

// ---------------------------------------------------------------------------
// Gather zq = emb[idx] (forward value of z_q_st) + per-(row,half) MSE.
// ---------------------------------------------------------------------------
__global__ __launch_bounds__(256)
void gather_mse_kernel(const float* __restrict__ h,
                       const float* __restrict__ emb1, const float* __restrict__ emb2,
                       const int* __restrict__ idx1, const int* __restrict__ idx2,
                       float* __restrict__ zq, float* __restrict__ mse) {
    const int half = blockIdx.y;
    const int row  = blockIdx.x;
    const float* emb = half ? emb2 : emb1;
    const int code = (half ? idx2 : idx1)[row];
    const int tT = threadIdx.x;

    float e = emb[(size_t)code * D_ + tT];
    float z = h[(size_t)row * TWO_D + half * D_ + tT];
    zq[(size_t)row * TWO_D + half * D_ + tT] = e;

    float d = e - z;
    __shared__ float s[256];
    s[tT] = d * d;
    __syncthreads();
    for (int off = 128; off > 0; off >>= 1) {
        if (tT < off) s[tT] += s[tT + off];
        __syncthreads();
    }
    if (tT == 0) mse[half * NROWS + row] = s[0] * (1.0f / D_);
}

// loss = 0.5 * (1+beta) * (mse1 + mse2) = 0.625 * (mse1 + mse2)
__global__ __launch_bounds__(256)
void loss_combine_kernel(const float* __restrict__ mse, float* __restrict__ loss) {
    int i = blockIdx.x * 256 + threadIdx.x;
    if (i < NROWS) loss[i] = 0.625f * (mse[i] + mse[NROWS + i]);
}

// ---------------------------------------------------------------------------
// Launch
// ---------------------------------------------------------------------------
extern "C" void kernel_launch(void* const* d_in, const int* in_sizes, int n_in,
                              void* d_out, int out_size, void* d_ws, size_t ws_size,
                              hipStream_t stream) {
    const float* z    = (const float*)d_in[0];
    const float* W_qa = (const float*)d_in[1];
    const float* b_qa = (const float*)d_in[2];
    const float* emb1 = (const float*)d_in[3];
    const float* emb2 = (const float*)d_in[4];
    const float* W_pq = (const float*)d_in[5];
    const float* b_pq = (const float*)d_in[6];

    // Output layout: z_q | idx1 | idx2 | loss (flat, return order)
    float* out_zq   = (float*)d_out;
    int*   out_idx1 = (int*)d_out + (size_t)NROWS * OUT_DIM;
    int*   out_idx2 = out_idx1 + NROWS;
    float* out_loss = (float*)((int*)d_out + (size_t)NROWS * OUT_DIM + 2 * NROWS);

    // Workspace: h | zq | enorm | mse  (~64.3 MB)
    float* ws    = (float*)d_ws;
    float* h     = ws;
    float* zq    = h + (size_t)NROWS * TWO_D;
    float* enorm = zq + (size_t)NROWS * TWO_D;
    float* mse   = enorm + 2 * KCODES;

    enorm_kernel<<<(2 * KCODES) / 256, 256, 0, stream>>>(emb1, emb2, enorm);

    gemm_bias_f32<<<dim3(TWO_D / 64, NROWS / 64), 128, 0, stream>>>(
        z, W_qa, b_qa, h, NROWS, TWO_D, IN_DIM);

    vq_argmin_kernel<<<dim3(NROWS / 64, 2), 128, 0, stream>>>(
        h, emb1, emb2, enorm, out_idx1, out_idx2);

    gather_mse_kernel<<<dim3(NROWS, 2), 256, 0, stream>>>(
        h, emb1, emb2, out_idx1, out_idx2, zq, mse);

    loss_combine_kernel<<<NROWS / 256, 256, 0, stream>>>(mse, out_loss);

    gemm_bias_f32<<<dim3(OUT_DIM / 64, NROWS / 64), 128, 0, stream>>>(
        zq, W_pq, b_pq, out_zq, NROWS, OUT_DIM, TWO_D);
}